// Help_36567351558250
// MI455X (gfx1250) — compile-verified
//
#include <hip/hip_runtime.h>
#include <stdint.h>

#define BLOCK 256

// Per-element Gaussian-splat covariance/view transform (bandwidth-bound streaming).
// CDNA5 path: view_matrix rows (64B/elem, dominant traffic) are DMA'd into LDS via
// global_load_async_to_lds_b128 (ASYNCcnt), overlapped with quaternion math on
// VGPR-loaded operands; outputs are transposed through LDS for coalesced b128 stores.
__global__ __launch_bounds__(BLOCK) void gs_cov_fwd(
    const float* __restrict__ rot,
    const float* __restrict__ mod,
    const float* __restrict__ scale,
    const float* __restrict__ p_k,
    const float* __restrict__ vm,
    float* __restrict__ out,
    int n)
{
    // 16 KB: phase A = view-matrix staging (256 * 64B), phase B = output transpose (256*9 f32)
    __shared__ float smem[BLOCK * 16];

    const int t   = threadIdx.x;
    const int i   = blockIdx.x * BLOCK + t;
    const int idx = (i < n) ? i : (n - 1);   // clamp tail: keep EXEC all-ones for async ops

    // ---------------- Phase 0: launch async DMA of this lane's 64B view-matrix row ----
    const uint32_t lds_base = (uint32_t)(uintptr_t)(&smem[0]);   // low 32b of flat LDS addr = LDS offset
    const uint32_t swz      = ((uint32_t)t & 3u) ^ (((uint32_t)t >> 2) & 3u); // bank de-conflict swizzle
    const unsigned long long gvm =
        (unsigned long long)(uintptr_t)vm + (unsigned long long)idx * 64ull;

#pragma unroll
    for (uint32_t c = 0; c < 4; ++c) {
        uint32_t ldst = lds_base + (uint32_t)t * 64u + (((c ^ swz) & 3u) << 4);
        unsigned long long ga = gvm + (unsigned long long)(c << 4);
        asm volatile("global_load_async_to_lds_b128 %0, %1, off"
                     :: "v"(ldst), "v"(ga)
                     : "memory");
    }

    // ---------------- Phase 1: small VGPR loads + quaternion math (overlaps DMA) ------
    const float4 q  = ((const float4*)rot)[idx];        // global_load_b128, coalesced
    const float2 sc = ((const float2*)scale)[idx];      // global_load_b64
    const float  m  = mod[0];                            // uniform -> SMEM load
    const float  p0 = p_k[3 * idx + 0];
    const float  p1 = p_k[3 * idx + 1];
    const float  p2 = p_k[3 * idx + 2];

    const float r = q.x, x = q.y, y = q.z, z = q.w;
    const float R00 = 1.0f - 2.0f * (y * y + z * z);
    const float R10 = 2.0f * (x * y + r * z);
    const float R20 = 2.0f * (x * z - r * y);
    const float R01 = 2.0f * (x * y - r * z);
    const float R11 = 1.0f - 2.0f * (x * x + z * z);
    const float R21 = 2.0f * (y * z + r * x);

    const float s0 = m * sc.x;
    const float s1 = m * sc.y;
    const float u0 = s0 * R00, u1 = s0 * R10, u2 = s0 * R20;
    const float v0 = s1 * R01, v1 = s1 * R11, v2 = s1 * R21;

    // ---------------- Phase 2: wait for DMA, consume view matrix from LDS -------------
    asm volatile("s_wait_asynccnt 0" ::: "memory");

    const int vbase = t * 16;
    const float4 vc0 = *(const float4*)&smem[vbase + (((0u ^ swz) & 3u) << 2)]; // vm[0..3]
    const float4 vc1 = *(const float4*)&smem[vbase + (((1u ^ swz) & 3u) << 2)]; // vm[4..7]
    const float4 vc2 = *(const float4*)&smem[vbase + (((2u ^ swz) & 3u) << 2)]; // vm[8..11]
    const float4 vc3 = *(const float4*)&smem[vbase + (((3u ^ swz) & 3u) << 2)]; // vm[12..15]

    const float r1s1 = vc0.x * u0 + vc1.x * u1 + vc2.x * u2;
    const float r1s2 = vc0.x * v0 + vc1.x * v1 + vc2.x * v2;
    const float r2s1 = vc0.y * u0 + vc1.y * u1 + vc2.y * u2;
    const float r2s2 = vc0.y * v0 + vc1.y * v1 + vc2.y * v2;
    const float r3s1 = vc0.z * u0 + vc1.z * u1 + vc2.z * u2;
    const float r3s2 = vc0.z * v0 + vc1.z * v1 + vc2.z * v2;
    const float r1p  = vc0.x * p0 + vc1.x * p1 + vc2.x * p2 + vc3.x;
    const float r2p  = vc0.y * p0 + vc1.y * p1 + vc2.y * p2 + vc3.y;
    const float r3p  = vc0.z * p0 + vc1.z * p1 + vc2.z * p2 + vc3.z;

    // ---------------- Phase 3: transpose outputs through LDS for coalesced stores -----
    __syncthreads();   // all lanes done reading staged vm before we overwrite smem

    const float a[9] = { r1s1, r1s2, r1p, r2s1, r2s2, r2p, r3s1, r3s2, r3p };
#pragma unroll
    for (int j = 0; j < 9; ++j)
        smem[t * 9 + j] = a[j];   // stride 9 dwords: 9 coprime with 64 banks -> conflict-free

    __syncthreads();

    // 2304 dwords per block, stored as coalesced b128: dwords [p*1024 + 4t .. +3]
    const size_t total = (size_t)n * 9;
    const size_t obase = (size_t)blockIdx.x * (size_t)(BLOCK * 9);
#pragma unroll
    for (int p = 0; p < 3; ++p) {
        const int d = p * 1024 + t * 4;
        if (d < BLOCK * 9) {
            const size_t o = obase + (size_t)d;
            if (o + 4 <= total) {
                *(float4*)(out + o) = *(const float4*)&smem[d];
            } else {
#pragma unroll
                for (int e = 0; e < 4; ++e)
                    if (o + (size_t)e < total) out[o + e] = smem[d + e];
            }
        }
    }
}

extern "C" void kernel_launch(void* const* d_in, const int* in_sizes, int n_in,
                              void* d_out, int out_size, void* d_ws, size_t ws_size,
                              hipStream_t stream) {
    (void)n_in; (void)out_size; (void)d_ws; (void)ws_size;
    const float* rot   = (const float*)d_in[0];   // (N,4)
    const float* mod   = (const float*)d_in[1];   // (1,)
    const float* scale = (const float*)d_in[2];   // (N,2)
    const float* p_k   = (const float*)d_in[3];   // (N,3)
    const float* vm    = (const float*)d_in[4];   // (N,16)
    float* out = (float*)d_out;                   // (N,9)

    const int n = in_sizes[0] / 4;
    const int blocks = (n + BLOCK - 1) / BLOCK;
    gs_cov_fwd<<<blocks, BLOCK, 0, stream>>>(rot, mod, scale, p_k, vm, out, n);
}